// SS3DVideoMamba_76252849373703
// MI455X (gfx1250) — compile-verified
//
#include <hip/hip_runtime.h>
#include <hip/hip_bf16.h>

typedef __attribute__((ext_vector_type(2))) float v2f;
typedef __attribute__((ext_vector_type(8))) float v8f;

#define BB 2
#define DD 8
#define HH 32
#define WW 32
#define LL 8192           // DD*HH*WW
#define DMODEL 96
#define DINNER 192
#define DSTATE 16
#define DTRANK 6
#define KDIR 6
#define CROWS 38          // DTRANK + 2*DSTATE

// ---- scan-order -> canonical spatial index (reversal folded in) ----
__device__ __forceinline__ int s_of_l(int k, int l) {
  int ll = (k & 1) ? (LL - 1 - l) : l;
  int kk = k >> 1;
  if (kk == 0) return ll;                       // (d,h,w) canonical
  if (kk == 1) {                                // l = d*W*H + w*H + h
    int d = ll >> 10; int r = ll & 1023;
    int w = r >> 5; int h = r & 31;
    return (d << 10) | (h << 5) | w;
  }
  // kk==2: l = h*W*Dd + w*Dd + d   (W*Dd = 256)
  int h = ll >> 8; int r = ll & 255;
  int w = r >> 3; int d = r & 7;
  return (d << 10) | (h << 5) | w;
}

// ================= generic C[M,N] = A[M,K] * W[N,K]^T (fp32 WMMA) =========
__global__ __launch_bounds__(128) void gemm_xwT(const float* __restrict__ A,
                                                const float* __restrict__ W,
                                                float* __restrict__ C,
                                                int M, int N, int Kd) {
  int lane = threadIdx.x & 31;
  int wave = threadIdx.x >> 5;
  int n0 = (blockIdx.x * 4 + wave) * 16;
  int m0 = blockIdx.y * 16;
  if (n0 >= N) return;                       // wave-uniform exit
  int half = lane >> 4, row = lane & 15;
  const float* arow = A + (size_t)(m0 + row) * Kd;
  const float* wrow = W + (size_t)(n0 + row) * Kd;
  v8f acc = {};
  for (int kk = 0; kk < Kd; kk += 4) {
    int kb = kk + 2 * half;
    v2f a; a.x = arow[kb]; a.y = arow[kb + 1];
    v2f b; b.x = wrow[kb]; b.y = wrow[kb + 1];
    acc = __builtin_amdgcn_wmma_f32_16x16x4_f32(false, a, false, b,
                                                (short)0, acc, false, false);
  }
  float* crow = C + (size_t)m0 * N + n0 + row;
#pragma unroll
  for (int v = 0; v < 8; ++v) {
    int m = v + 8 * half;
    crow[(size_t)m * N] = acc[v];
  }
}

// ============== depthwise 3x3x3 conv + bias + SiLU =========================
__global__ __launch_bounds__(256) void dwconv_silu(const float* __restrict__ xz,
                                                   const float* __restrict__ cw,
                                                   const float* __restrict__ cb,
                                                   float* __restrict__ uc) {
  int tid = blockIdx.x * blockDim.x + threadIdx.x;
  if (tid >= BB * LL * DINNER) return;
  int c = tid % DINNER;
  int p = tid / DINNER;             // b*L + s
  int b = p / LL; int s = p % LL;
  int d = s >> 10, h = (s >> 5) & 31, w = s & 31;
  float acc = cb[c];
  const float* wt = cw + c * 27;
#pragma unroll
  for (int kd = 0; kd < 3; ++kd) {
    int dd = d + kd - 1;
    if ((unsigned)dd >= DD) continue;
#pragma unroll
    for (int kh = 0; kh < 3; ++kh) {
      int hh = h + kh - 1;
      if ((unsigned)hh >= HH) continue;
#pragma unroll
      for (int kw = 0; kw < 3; ++kw) {
        int ww2 = w + kw - 1;
        if ((unsigned)ww2 >= WW) continue;
        int sp = (dd << 10) | (hh << 5) | ww2;
        acc += xz[((size_t)(b * LL + sp)) * (2 * DINNER) + c] * wt[kd * 9 + kh * 3 + kw];
      }
    }
  }
  uc[tid] = acc / (1.f + __expf(-acc));
}

// ===== x_dbl[b,k,c,l] = sum_d xpw[k,c,d] * u_conv[b, s_k(l), d]  (WMMA) ====
__global__ __launch_bounds__(128) void xdbl_gemm(const float* __restrict__ uc,
                                                 const float* __restrict__ xpw,
                                                 float* __restrict__ xdbl) {
  int bk = blockIdx.z;                 // 0..11
  int b = bk / KDIR, k = bk % KDIR;
  int lane = threadIdx.x & 31;
  int wave = threadIdx.x >> 5;
  int m0 = blockIdx.y * 16;            // 0,16,32 (covers 38 rows w/ masking)
  int l0 = (blockIdx.x * 4 + wave) * 16;
  int half = lane >> 4, row = lane & 15;
  int m = m0 + row;
  float amask = (m < CROWS) ? 1.f : 0.f;
  const float* arow = xpw + ((size_t)k * CROWS + (m < CROWS ? m : CROWS - 1)) * DINNER;
  int s = s_of_l(k, l0 + row);
  const float* brow = uc + ((size_t)b * LL + s) * DINNER;
  v8f acc = {};
  for (int kk = 0; kk < DINNER; kk += 4) {
    int kb = kk + 2 * half;
    v2f a; a.x = arow[kb] * amask; a.y = arow[kb + 1] * amask;
    v2f bfr; bfr.x = brow[kb]; bfr.y = brow[kb + 1];
    acc = __builtin_amdgcn_wmma_f32_16x16x4_f32(false, a, false, bfr,
                                                (short)0, acc, false, false);
  }
  float* base = xdbl + (size_t)bk * CROWS * LL;
#pragma unroll
  for (int v = 0; v < 8; ++v) {
    int mm = m0 + v + 8 * half;
    if (mm < CROWS) base[(size_t)mm * LL + l0 + row] = acc[v];
  }
}

// ===================== selective scan, state-parallel ======================
// block: 1024 threads = (n in 0..15) x (64 channels); grid (3, 12)
// Double-buffered async global->LDS staging of the 38x32 x_dbl chunk:
// issue chunk c+1 with GLOBAL_LOAD_ASYNC_TO_LDS_B64 while computing chunk c.
#define CHUNK 32
#define CELEM (CROWS * CHUNK)          // 1216 floats per chunk
#define CPAIR (CELEM / 2)              // 608 b64 copies

__global__ __launch_bounds__(1024) void scan_kernel(const float* __restrict__ uc,
                                                    const float* __restrict__ xdbl,
                                                    const float* __restrict__ dtw,
                                                    const float* __restrict__ dtb,
                                                    const float* __restrict__ Alogs,
                                                    float* __restrict__ ys) {
  int bk = blockIdx.y; int b = bk / KDIR, k = bk % KDIR;
  int t = threadIdx.x;
  int n = t & 15;
  int d = blockIdx.x * 64 + (t >> 4);
  __shared__ float sh[2][CELEM];
  const float* wr = dtw + ((size_t)k * DINNER + d) * DTRANK;
  float w0 = wr[0], w1 = wr[1], w2 = wr[2], w3 = wr[3], w4 = wr[4], w5 = wr[5];
  float bias = dtb[k * DINNER + d];
  float Aa = -__expf(Alogs[((size_t)(k * DINNER + d)) * DSTATE + n]);
  float h = 0.f;
  const float* xdbase = xdbl + (size_t)bk * CROWS * LL;
  const float* ucb = uc + (size_t)b * LL * DINNER;
  float* ysb = ys + (size_t)bk * LL * DINNER;

  // async-issue helper state (per-thread constants)
  int e = t * 2;                       // element pair index for staging
  int erow = e >> 5, ecol = e & 31;    // row/col within 38x32 chunk

  // prologue: stage chunk 0 into buffer 0
  if (t < CPAIR) {
    const float* g = xdbase + (size_t)erow * LL + ecol;
    unsigned lo = (unsigned)(uintptr_t)&sh[0][e];
    asm volatile("global_load_async_to_lds_b64 %0, %1, off"
                 :: "v"(lo), "v"(g) : "memory");
  }

  int buf = 0;
  for (int l0 = 0; l0 < LL; l0 += CHUNK) {
    // wait for chunk in `buf` (each issuing wave waits its own in-order op)
    asm volatile("s_wait_asynccnt 0x0" ::: "memory");
    __syncthreads();                   // chunk resident for all waves
    // kick off next chunk into the other buffer (overlaps with compute)
    if (l0 + CHUNK < LL && t < CPAIR) {
      const float* g = xdbase + (size_t)erow * LL + (l0 + CHUNK) + ecol;
      unsigned lo = (unsigned)(uintptr_t)&sh[buf ^ 1][e];
      asm volatile("global_load_async_to_lds_b64 %0, %1, off"
                   :: "v"(lo), "v"(g) : "memory");
    }
    const float* S = sh[buf];
#pragma unroll 4
    for (int tt = 0; tt < CHUNK; ++tt) {
      int l = l0 + tt;
      int s = s_of_l(k, l);
      float u = ucb[(size_t)s * DINNER + d];
      float dtv = bias + w0 * S[0 * 32 + tt] + w1 * S[1 * 32 + tt]
                       + w2 * S[2 * 32 + tt] + w3 * S[3 * 32 + tt]
                       + w4 * S[4 * 32 + tt] + w5 * S[5 * 32 + tt];
      float delta = (dtv > 20.f) ? dtv : log1pf(__expf(dtv));
      float du = delta * u;
      h = __expf(delta * Aa) * h + du * S[(6 + n) * 32 + tt];
      float y = h * S[(22 + n) * 32 + tt];
      y += __shfl_xor(y, 1);
      y += __shfl_xor(y, 2);
      y += __shfl_xor(y, 4);
      y += __shfl_xor(y, 8);
      if (n == 0) ysb[(size_t)l * DINNER + d] = y;
    }
    __syncthreads();                   // done reading `buf` before chunk c+2 lands
    buf ^= 1;
  }
}

// ============ merge 6 directions + D*u + LayerNorm + SiLU gate =============
__global__ __launch_bounds__(192) void combine_ln_gate(const float* __restrict__ ys,
                                                       const float* __restrict__ uc,
                                                       const float* __restrict__ xz,
                                                       const float* __restrict__ Ds,
                                                       const float* __restrict__ gamma,
                                                       const float* __restrict__ beta,
                                                       float* __restrict__ yg) {
  int p = blockIdx.x;            // b*L + s
  int c = threadIdx.x;           // 0..191
  int b = p / LL; int s = p % LL;
  int d = s >> 10, h = (s >> 5) & 31, w = s & 31;
  int lx = s;
  int ly = (d << 10) | (w << 5) | h;          // d*W*H + w*H + h
  int lz = (h << 8) | (w << 3) | d;           // h*W*Dd + w*Dd + d
  int larr[6] = { lx, LL - 1 - lx, ly, LL - 1 - ly, lz, LL - 1 - lz };
  float val = 0.f, dsum = 0.f;
#pragma unroll
  for (int k = 0; k < KDIR; ++k) {
    val += ys[(((size_t)b * KDIR + k) * LL + larr[k]) * DINNER + c];
    dsum += Ds[k * DINNER + c];
  }
  val += dsum * uc[(size_t)p * DINNER + c];

  __shared__ float s1[DINNER], s2[DINNER], mvs[2];
  s1[c] = val; s2[c] = val * val;
  __syncthreads();
  for (int off = 96; off >= 3; off >>= 1) {
    if (c < off) { s1[c] += s1[c + off]; s2[c] += s2[c + off]; }
    __syncthreads();
  }
  if (c == 0) {
    float sum = s1[0] + s1[1] + s1[2];
    float sq  = s2[0] + s2[1] + s2[2];
    float mu = sum * (1.f / DINNER);
    mvs[0] = mu;
    mvs[1] = sq * (1.f / DINNER) - mu * mu;
  }
  __syncthreads();
  float ynorm = (val - mvs[0]) * rsqrtf(mvs[1] + 1e-5f) * gamma[c] + beta[c];
  float z = xz[(size_t)p * (2 * DINNER) + DINNER + c];
  yg[(size_t)p * DINNER + c] = ynorm * (z / (1.f + __expf(-z)));
}

// ===========================================================================
extern "C" void kernel_launch(void* const* d_in, const int* in_sizes, int n_in,
                              void* d_out, int out_size, void* d_ws, size_t ws_size,
                              hipStream_t stream) {
  const float* x        = (const float*)d_in[0];   // (B,Dd,H,W,96)
  const float* inw      = (const float*)d_in[1];   // (384,96)
  const float* convw    = (const float*)d_in[2];   // (192,1,3,3,3)
  const float* convb    = (const float*)d_in[3];   // (192,)
  const float* xpw      = (const float*)d_in[4];   // (6,38,192)
  const float* dtw      = (const float*)d_in[5];   // (6,192,6)
  const float* dtb      = (const float*)d_in[6];   // (6,192)
  const float* Alogs    = (const float*)d_in[7];   // (1152,16)
  const float* Ds       = (const float*)d_in[8];   // (1152,)
  const float* gamma    = (const float*)d_in[9];   // (192,)
  const float* beta     = (const float*)d_in[10];  // (192,)
  const float* opw      = (const float*)d_in[11];  // (96,192)
  float* out = (float*)d_out;

  float* ws   = (float*)d_ws;
  float* xz   = ws;                       // 16384*384  = 6,291,456
  float* ucv  = xz   + (size_t)16384 * 384;   // 16384*192
  float* xdbl = ucv  + (size_t)16384 * 192;   // 12*38*8192
  float* ysb  = xdbl + (size_t)12 * CROWS * LL; // 12*8192*192
  float* yg   = ysb  + (size_t)12 * LL * DINNER;

  // 1) in_proj: xz[16384,384] = x[16384,96] @ inw^T
  gemm_xwT<<<dim3(384 / 64, 16384 / 16), 128, 0, stream>>>(x, inw, xz, 16384, 384, 96);
  // 2) depthwise conv + SiLU -> u_conv[b,s,c]
  dwconv_silu<<<(BB * LL * DINNER + 255) / 256, 256, 0, stream>>>(xz, convw, convb, ucv);
  // 3) x_dbl[b,k,c,l] (scan-order permutation folded into B addressing)
  xdbl_gemm<<<dim3(LL / 64, 3, 12), 128, 0, stream>>>(ucv, xpw, xdbl);
  // 4) selective scan, state-parallel, async double-buffered staging
  scan_kernel<<<dim3(3, 12), 1024, 0, stream>>>(ucv, xdbl, dtw, dtb, Alogs, ysb);
  // 5) merge + LayerNorm + gate
  combine_ln_gate<<<BB * LL, DINNER, 0, stream>>>(ysb, ucv, xz, Ds, gamma, beta, yg);
  // 6) out_proj: out[16384,96] = yg[16384,192] @ opw^T
  gemm_xwT<<<dim3(2, 16384 / 16), 128, 0, stream>>>(yg, opw, out, 16384, 96, 192);
}